// DeformerNet_44049184587989
// MI455X (gfx1250) — compile-verified
//
#include <hip/hip_runtime.h>
#include <hip/hip_bf16.h>
#include <math.h>
#include <stdint.h>

// ---------------------------------------------------------------------------
// DeformerNet on MI455X (gfx1250, wave32).  fp32 end-to-end, all matrix ops
// routed through V_WMMA_F32_16X16X4_F32 (native fp32 WMMA, K=4 steps).
// Point arrays staged to LDS via GLOBAL_LOAD_ASYNC_TO_LDS_B128 (ASYNCcnt).
// ---------------------------------------------------------------------------

typedef __attribute__((ext_vector_type(2))) float v2f;
typedef __attribute__((ext_vector_type(8))) float v8f;

#define BATCH 8
#define NPTS  2048
#define EPSF  1e-5f

__device__ __forceinline__ v8f wmma_f32(v2f a, v2f b, v8f c) {
  // 8 args: (neg_a, A, neg_b, B, c_mod, C, reuse_a, reuse_b)
  return __builtin_amdgcn_wmma_f32_16x16x4_f32(false, a, false, b, (short)0, c,
                                               false, false);
}

// CDNA5 async global->LDS copy (16B), tracked by ASYNCcnt.
__device__ __forceinline__ void async_copy_b128(uint32_t lds_byte_off,
                                                const void* gptr) {
  asm volatile("global_load_async_to_lds_b128 %0, %1, off"
               :
               : "v"(lds_byte_off), "v"((uint64_t)(uintptr_t)gptr)
               : "memory");
}
__device__ __forceinline__ void wait_async0() {
  asm volatile("s_wait_asynccnt 0x0" ::: "memory");
}

static __host__ __device__ __forceinline__ int ru4(int x) { return (x + 3) & ~3; }

// ---------------------------------------------------------------------------
// Utility kernels
// ---------------------------------------------------------------------------
__global__ void zero_f32(float* __restrict__ p, int n) {
  int i = blockIdx.x * blockDim.x + threadIdx.x;
  if (i < n) p[i] = 0.f;
}

// xyz (B,3,N) channel-first  ->  pts (B,N,3) channel-last
__global__ void to_chlast(const float* __restrict__ xyz, float* __restrict__ pts,
                          int n) {
  int b = blockIdx.y;
  int i = blockIdx.x * blockDim.x + threadIdx.x;
  if (i < n) {
    for (int c = 0; c < 3; ++c)
      pts[((size_t)b * n + i) * 3 + c] = xyz[((size_t)b * 3 + c) * n + i];
  }
}

// ---------------------------------------------------------------------------
// Density kernel: one wave computes a 16-row block of the NxN Gram matrix
// via WMMA 16x16x4 (K = 3 coords, zero-padded to 4), accumulating
// sum_j exp(-d2/(2 bw^2)) per row.  inv_d = 1/density.
// ---------------------------------------------------------------------------
__global__ void density_kernel(const float* __restrict__ pts, /* (B,n,3) */
                               float* __restrict__ inv_d, int n, float bw) {
  int b = blockIdx.y;
  int wave = threadIdx.x >> 5;
  int rt = (blockIdx.x * 4 + wave) * 16;  // row-tile base
  int lane = threadIdx.x & 31;
  bool hi = lane >= 16;
  int l = lane & 15;
  const float* P = pts + (size_t)b * n * 3;

  // my staged row (rt+l) coords and |x|^2
  float rx = P[(size_t)(rt + l) * 3 + 0];
  float ry = P[(size_t)(rt + l) * 3 + 1];
  float rz = P[(size_t)(rt + l) * 3 + 2];
  float n2r = rx * rx + ry * ry + rz * rz;

  v2f a;
  if (!hi) { a.x = rx; a.y = ry; } else { a.x = rz; a.y = 0.f; }

  float inv2bw2 = 1.f / (2.f * bw * bw);
  float accs[8];
#pragma unroll
  for (int v = 0; v < 8; ++v) accs[v] = 0.f;

  for (int ct = 0; ct < n; ct += 16) {
    float cx = P[(size_t)(ct + l) * 3 + 0];
    float cy = P[(size_t)(ct + l) * 3 + 1];
    float cz = P[(size_t)(ct + l) * 3 + 2];
    float n2c = cx * cx + cy * cy + cz * cz;  // col = ct + l (both halves)
    v2f bf;
    if (!hi) { bf.x = cx; bf.y = cy; } else { bf.x = cz; bf.y = 0.f; }
    v8f g = {};
    g = wmma_f32(a, bf, g);  // G[i,j] = x_i . x_j for the 16x16 tile
#pragma unroll
    for (int v = 0; v < 8; ++v) {
      int rloc = v + (hi ? 8 : 0);
      float n2row = __shfl(n2r, rloc, 32);
      float d2 = n2row + n2c - 2.f * g[v];
      accs[v] += __expf(-d2 * inv2bw2);
    }
  }
  // reduce each row-sum across the 16 lanes holding its columns
#pragma unroll
  for (int v = 0; v < 8; ++v) {
    float s = accs[v];
    s += __shfl_xor(s, 1, 32);
    s += __shfl_xor(s, 2, 32);
    s += __shfl_xor(s, 4, 32);
    s += __shfl_xor(s, 8, 32);
    if (l == 0) {
      int row = rt + v + (hi ? 8 : 0);
      // density = (s/n) / (2.5*bw)  ->  inv = 2.5*bw*n / s
      inv_d[(size_t)b * n + row] = 2.5f * bw * (float)n / s;
    }
  }
}

// ---------------------------------------------------------------------------
// Furthest point sampling: one block per batch, sequential over npoint iters.
// Point array staged to LDS with async b128 copies; the 512-iteration scan
// then runs entirely out of LDS.
// ---------------------------------------------------------------------------
__global__ void fps_kernel(const float* __restrict__ pts, int n, int npoint,
                           int* __restrict__ fps_idx,
                           float* __restrict__ new_xyz /* (B,npoint,3) */) {
  __shared__ float spts[2048 * 3];
  __shared__ float dist[2048];
  __shared__ float rv[256];
  __shared__ int ri[256];
  __shared__ int sfar;
  __shared__ float cen[3];
  int b = blockIdx.x, t = threadIdx.x;
  const float* P = pts + (size_t)b * n * 3;

  // async stage (n*3 floats, 16B chunks) into LDS
  {
    uint32_t lbase = (uint32_t)(uintptr_t)&spts[0];
    int chunks = (n * 3) / 4;
    for (int i = t; i < chunks; i += 256)
      async_copy_b128(lbase + (uint32_t)i * 16u, P + (size_t)i * 4);
    wait_async0();
  }
  for (int i = t; i < n; i += 256) dist[i] = 1e10f;
  if (t == 0) sfar = 0;
  __syncthreads();

  for (int it = 0; it < npoint; ++it) {
    int far = sfar;
    if (t == 0) {
      fps_idx[b * npoint + it] = far;
      cen[0] = spts[far * 3 + 0];
      cen[1] = spts[far * 3 + 1];
      cen[2] = spts[far * 3 + 2];
      new_xyz[((size_t)b * npoint + it) * 3 + 0] = cen[0];
      new_xyz[((size_t)b * npoint + it) * 3 + 1] = cen[1];
      new_xyz[((size_t)b * npoint + it) * 3 + 2] = cen[2];
    }
    __syncthreads();
    float bm = -1.f;
    int bi = 0;
    for (int i = t; i < n; i += 256) {
      float dx = spts[i * 3 + 0] - cen[0];
      float dy = spts[i * 3 + 1] - cen[1];
      float dz = spts[i * 3 + 2] - cen[2];
      float d = dx * dx + dy * dy + dz * dz;
      float nd = fminf(dist[i], d);
      dist[i] = nd;
      if (nd > bm) { bm = nd; bi = i; }
    }
    rv[t] = bm; ri[t] = bi;
    __syncthreads();
    for (int off = 128; off; off >>= 1) {
      if (t < off) {
        if (rv[t + off] > rv[t] ||
            (rv[t + off] == rv[t] && ri[t + off] < ri[t])) {
          rv[t] = rv[t + off];
          ri[t] = ri[t + off];
        }
      }
      __syncthreads();
    }
    if (t == 0) sfar = ri[0];
    __syncthreads();
  }
}

// ---------------------------------------------------------------------------
// kNN: one wave per (b,s) centroid; points async-staged to LDS, then nsample
// iterative argmin passes over an LDS-resident distance array.
// ---------------------------------------------------------------------------
__global__ void knn_kernel(const float* __restrict__ pts,
                           const float* __restrict__ new_xyz, int n, int S,
                           int nsample, int* __restrict__ nn_idx) {
  __shared__ float spts[2048 * 3];
  __shared__ float d2s[2048];
  int b = blockIdx.y, s = blockIdx.x;
  int lane = threadIdx.x;
  const float* P = pts + (size_t)b * n * 3;
  size_t bs = (size_t)b * S + s;
  float cx = new_xyz[bs * 3 + 0];
  float cy = new_xyz[bs * 3 + 1];
  float cz = new_xyz[bs * 3 + 2];
  {
    uint32_t lbase = (uint32_t)(uintptr_t)&spts[0];
    int chunks = (n * 3) / 4;
    for (int i = lane; i < chunks; i += 32)
      async_copy_b128(lbase + (uint32_t)i * 16u, P + (size_t)i * 4);
    wait_async0();
  }
  __syncthreads();
  for (int i = lane; i < n; i += 32) {
    float dx = spts[i * 3 + 0] - cx;
    float dy = spts[i * 3 + 1] - cy;
    float dz = spts[i * 3 + 2] - cz;
    d2s[i] = dx * dx + dy * dy + dz * dz;
  }
  __syncthreads();
  for (int k = 0; k < nsample; ++k) {
    float best = 3.4e38f;
    int bi = 0;
    for (int i = lane; i < n; i += 32) {
      float d = d2s[i];
      if (d < best) { best = d; bi = i; }
    }
    for (int off = 16; off; off >>= 1) {
      float ob = __shfl_xor(best, off, 32);
      int oi = __shfl_xor(bi, off, 32);
      if (ob < best || (ob == best && oi < bi)) { best = ob; bi = oi; }
    }
    if (lane == 0) {
      nn_idx[bs * nsample + k] = bi;
      d2s[bi] = 3.4e38f;
    }
    __syncthreads();
  }
}

// ---------------------------------------------------------------------------
// Grouping: builds row-major inputs for the three MLP chains.
// ---------------------------------------------------------------------------
__global__ void group_kernel(const float* __restrict__ pts,
                             const float* __restrict__ feat, int Cf,
                             const float* __restrict__ inv_d,
                             const float* __restrict__ new_xyz,
                             const int* __restrict__ nn_idx, int n, int S,
                             int Ksamp, float* __restrict__ Xmlp, int KPm,
                             float* __restrict__ Xw, float* __restrict__ Xds) {
  __shared__ float gden[128];
  int b = blockIdx.y, s = blockIdx.x;
  int k = threadIdx.x;
  size_t bs = (size_t)b * S + s;
  int idx = nn_idx ? nn_idx[bs * Ksamp + k] : k;
  float nx = 0.f, ny = 0.f, nz = 0.f;
  if (new_xyz) {
    nx = new_xyz[bs * 3 + 0];
    ny = new_xyz[bs * 3 + 1];
    nz = new_xyz[bs * 3 + 2];
  }
  const float* p = pts + ((size_t)b * n + idx) * 3;
  float gx = p[0] - nx, gy = p[1] - ny, gz = p[2] - nz;
  size_t r = bs * Ksamp + k;
  float* xm = Xmlp + r * KPm;
  xm[0] = gx; xm[1] = gy; xm[2] = gz;
  const float* f = feat + ((size_t)b * n + idx) * Cf;
  for (int c = 0; c < Cf; ++c) xm[3 + c] = f[c];
  for (int c = 3 + Cf; c < KPm; ++c) xm[c] = 0.f;
  float* xw = Xw + r * 4;
  xw[0] = gx; xw[1] = gy; xw[2] = gz; xw[3] = 0.f;
  float gd = inv_d[(size_t)b * n + idx];
  gden[k] = gd;
  __syncthreads();
  for (int off = blockDim.x / 2; off; off >>= 1) {
    if (k < off) gden[k] = fmaxf(gden[k], gden[k + off]);
    __syncthreads();
  }
  float mx = gden[0];
  float* xd = Xds + r * 4;
  xd[0] = gd / mx; xd[1] = 0.f; xd[2] = 0.f; xd[3] = 0.f;
}

// ---------------------------------------------------------------------------
// WMMA tile GEMM:  Y[rows x ldy] = X[rows x ldx] . W^T (+ bias)
// CT = column tiles per wave (A fragment reused CT times -> 4x WMMA/A-load).
// Epilogue accumulates per-output-channel sum/sumsq into stats[2*O] for the
// reference's batch-stat BN.
// ---------------------------------------------------------------------------
template <int CT>
__global__ void gemm_bn_stats(const float* __restrict__ X, int ldx,
                              const float* __restrict__ W, int Kw,
                              const float* __restrict__ bias,
                              float* __restrict__ Y, int ldy, int rows, int KP,
                              int O, float* __restrict__ stats) {
  int rt = blockIdx.x * 16;
  int ctBase = blockIdx.y * (16 * CT);
  int lane = threadIdx.x;
  bool hi = lane >= 16;
  int l = lane & 15;
  int ar = rt + l;
  v8f acc[CT];
#pragma unroll
  for (int t = 0; t < CT; ++t) acc[t] = (v8f){};
  for (int k0 = 0; k0 < KP; k0 += 4) {
    v2f a;
    if (ar < rows) {
      const float* xr = X + (size_t)ar * ldx + k0 + (hi ? 2 : 0);
      a.x = xr[0];
      a.y = xr[1];
    } else {
      a.x = 0.f; a.y = 0.f;
    }
#pragma unroll
    for (int t = 0; t < CT; ++t) {
      int o = ctBase + t * 16 + l;
      v2f bfr;
      if (o < O) {
        int kA = k0 + (hi ? 2 : 0);
        int kB = kA + 1;
        const float* wr = W + (size_t)o * Kw;
        bfr.x = (kA < Kw) ? wr[kA] : 0.f;
        bfr.y = (kB < Kw) ? wr[kB] : 0.f;
        if (kA + 64 < Kw) __builtin_prefetch(&wr[kA + 64]);
      } else {
        bfr.x = 0.f; bfr.y = 0.f;
      }
      acc[t] = wmma_f32(a, bfr, acc[t]);
    }
  }
#pragma unroll
  for (int t = 0; t < CT; ++t) {
    int o = ctBase + t * 16 + l;
    float bv = (o < O) ? bias[o] : 0.f;
    float s = 0.f, s2 = 0.f;
#pragma unroll
    for (int v = 0; v < 8; ++v) {
      int row = rt + v + (hi ? 8 : 0);
      float y = acc[t][v] + bv;
      if (row < rows && o < ldy) Y[(size_t)row * ldy + o] = (o < O) ? y : 0.f;
      if (row < rows && o < O) { s += y; s2 += y * y; }
    }
    if (stats) {
      s += __shfl_xor(s, 16, 32);
      s2 += __shfl_xor(s2, 16, 32);
      if (!hi && o < O) {
        atomicAdd(&stats[o], s);
        atomicAdd(&stats[o + O], s2);
      }
    }
  }
}

__global__ void bn_finalize(const float* __restrict__ stats,
                            const float* __restrict__ g,
                            const float* __restrict__ be,
                            float* __restrict__ ss, int O, float inv_cnt) {
  int c = blockIdx.x * blockDim.x + threadIdx.x;
  if (c >= O) return;
  float m = stats[c] * inv_cnt;
  float var = stats[c + O] * inv_cnt - m * m;
  float sc = g[c] * rsqrtf(var + EPSF);
  ss[c] = sc;
  ss[c + O] = be[c] - m * sc;
}

__global__ void bn_apply_relu(float* __restrict__ Y, int ldy, int rows, int O,
                              const float* __restrict__ ss) {
  size_t i = (size_t)blockIdx.x * blockDim.x + threadIdx.x;
  size_t total = (size_t)rows * ldy;
  if (i >= total) return;
  int c = (int)(i % (size_t)ldy);
  if (c < O) {
    float y = Y[i] * ss[c] + ss[c + O];
    Y[i] = y > 0.f ? y : 0.f;
  }
}

// ---------------------------------------------------------------------------
// Pooling einsum 'bskc,bskw->bscw' with h scaled by ds:
//   pooled[bs, c*16 + w] = sum_k (H[bs,k,c] * ds[bs,k]) * Wn[bs,k,w]
// One wave per (bs, 64-channel group): the weightnet B fragment is shared
// across the 4 channel tiles (4 WMMA per B-load).
// ---------------------------------------------------------------------------
__global__ void pool_kernel(const float* __restrict__ H, int ldh,
                            const float* __restrict__ Wn, int ldw,
                            const float* __restrict__ ds, int Ksamp, int C,
                            float* __restrict__ pooled) {
  int bs = blockIdx.x;
  int c0 = blockIdx.y * 64;
  int lane = threadIdx.x;
  bool hi = lane >= 16;
  int l = lane & 15;
  size_t base = (size_t)bs * Ksamp;
  v8f acc[4];
#pragma unroll
  for (int t = 0; t < 4; ++t) acc[t] = (v8f){};
  for (int k0 = 0; k0 < Ksamp; k0 += 4) {
    int ka = k0 + (hi ? 2 : 0);
    float dsa = ds[(base + ka) * 4];
    float dsb = ds[(base + ka + 1) * 4];
    v2f bf;
    bf.x = Wn[(base + ka) * ldw + l];
    bf.y = Wn[(base + ka + 1) * ldw + l];
#pragma unroll
    for (int t = 0; t < 4; ++t) {
      int m = c0 + t * 16 + l;
      v2f a;
      a.x = H[(base + ka) * ldh + m] * dsa;
      a.y = H[(base + ka + 1) * ldh + m] * dsb;
      acc[t] = wmma_f32(a, bf, acc[t]);
    }
  }
#pragma unroll
  for (int t = 0; t < 4; ++t) {
#pragma unroll
    for (int v = 0; v < 8; ++v) {
      int c = c0 + t * 16 + v + (hi ? 8 : 0);
      pooled[(size_t)bs * (C * 16) + (size_t)c * 16 + l] = acc[t][v];
    }
  }
}

// ---------------------------------------------------------------------------
// Head: FC 512->256->128->64->9 with per-sample GroupNorm + relu, then
// pos / rotation construction.  One block per sample (8 blocks, tiny).
// ---------------------------------------------------------------------------
__device__ void gn_relu(float* v, int n, const float* __restrict__ g,
                        const float* __restrict__ b, float* red, int t,
                        int nt) {
  float s = 0.f;
  for (int i = t; i < n; i += nt) s += v[i];
  red[t] = s;
  __syncthreads();
  for (int off = nt / 2; off; off >>= 1) {
    if (t < off) red[t] += red[t + off];
    __syncthreads();
  }
  float m = red[0] / n;
  __syncthreads();
  float s2 = 0.f;
  for (int i = t; i < n; i += nt) {
    float d = v[i] - m;
    s2 += d * d;
  }
  red[t] = s2;
  __syncthreads();
  for (int off = nt / 2; off; off >>= 1) {
    if (t < off) red[t] += red[t + off];
    __syncthreads();
  }
  float inv = rsqrtf(red[0] / n + EPSF);
  __syncthreads();
  for (int i = t; i < n; i += nt) {
    float y = g[i] * (v[i] - m) * inv + b[i];
    v[i] = y > 0.f ? y : 0.f;
  }
  __syncthreads();
}

__global__ void head_kernel(
    const float* __restrict__ f0, const float* __restrict__ f1,
    const float* fc1_w, const float* fc1_b, const float* gn1_g,
    const float* gn1_b, const float* fc3_w, const float* fc3_b,
    const float* gn3_g, const float* gn3_b, const float* fc4_w,
    const float* fc4_b, const float* gn4_g, const float* gn4_b,
    const float* fc5_w, const float* fc5_b, float* __restrict__ out) {
  __shared__ float x[512];
  __shared__ float h1[256];
  __shared__ float h2[128];
  __shared__ float h3[64];
  __shared__ float x9[9];
  __shared__ float red[256];
  int b = blockIdx.x, t = threadIdx.x;
  for (int i = t; i < 256; i += 256) {
    x[i] = f0[b * 256 + i];
    x[256 + i] = f1[b * 256 + i];
  }
  __syncthreads();
  {
    float acc = fc1_b[t];
    const float* w = fc1_w + (size_t)t * 512;
    for (int k = 0; k < 512; ++k) acc += w[k] * x[k];
    h1[t] = acc;
  }
  __syncthreads();
  gn_relu(h1, 256, gn1_g, gn1_b, red, t, 256);
  if (t < 128) {
    float acc = fc3_b[t];
    const float* w = fc3_w + (size_t)t * 256;
    for (int k = 0; k < 256; ++k) acc += w[k] * h1[k];
    h2[t] = acc;
  }
  __syncthreads();
  gn_relu(h2, 128, gn3_g, gn3_b, red, t, 256);
  if (t < 64) {
    float acc = fc4_b[t];
    const float* w = fc4_w + (size_t)t * 128;
    for (int k = 0; k < 128; ++k) acc += w[k] * h2[k];
    h3[t] = acc;
  }
  __syncthreads();
  gn_relu(h3, 64, gn4_g, gn4_b, red, t, 256);
  if (t < 9) {
    float acc = fc5_b[t];
    const float* w = fc5_w + (size_t)t * 64;
    for (int k = 0; k < 64; ++k) acc += w[k] * h3[k];
    x9[t] = acc;
  }
  __syncthreads();
  if (t == 0) {
    float a1x = x9[3], a1y = x9[4], a1z = x9[5];
    float n1 = a1x * a1x + a1y * a1y + a1z * a1z;
    float r1 = rsqrtf(fmaxf(n1, 1e-8f));
    a1x *= r1; a1y *= r1; a1z *= r1;
    float vx = x9[6], vy = x9[7], vz = x9[8];
    float c3x = a1y * vz - a1z * vy;
    float c3y = a1z * vx - a1x * vz;
    float c3z = a1x * vy - a1y * vx;
    float n3 = c3x * c3x + c3y * c3y + c3z * c3z;
    float r3 = rsqrtf(fmaxf(n3, 1e-8f));
    c3x *= r3; c3y *= r3; c3z *= r3;
    float a2x = c3y * a1z - c3z * a1y;
    float a2y = c3z * a1x - c3x * a1z;
    float a2z = c3x * a1y - c3y * a1x;
    out[b * 3 + 0] = x9[0];
    out[b * 3 + 1] = x9[1];
    out[b * 3 + 2] = x9[2];
    float* R = out + BATCH * 3 + b * 9;  // rot[b][r][c], col c = a_{c+1}
    R[0] = a1x; R[1] = a2x; R[2] = c3x;
    R[3] = a1y; R[4] = a2y; R[5] = c3y;
    R[6] = a1z; R[7] = a2z; R[8] = c3z;
  }
}

// ---------------------------------------------------------------------------
// Host-side orchestration
// ---------------------------------------------------------------------------
struct SaP {
  const float *mlp_w, *mlp_b, *mlp_g, *mlp_be;
  const float *wn_w[3], *wn_b[3], *wn_g[3], *wn_be[3];
  const float *dn_w[3], *dn_b[3], *dn_g[3], *dn_be[3];
  const float *lin_w, *lin_b, *lin_g, *lin_be;
};

struct Bufs {
  float *pts0, *nx1, *nx2, *invd;
  float *Xmlp, *Xw, *Xds, *H;
  float *Wb1, *Wb2, *WbF, *Db1, *Db2, *DbF;
  float *pooled, *f1, *f2, *f3a, *f3b, *stats, *ss;
  int *fpsidx, *nnidx;
};

static void mlp_layer(hipStream_t st, const float* X, int ldx, int K,
                      const float* W, const float* b, const float* g,
                      const float* be, float* Y, int O, int rows, float* stats,
                      float* ss) {
  int KP = ru4(K);
  int ldy = ru4(O);
  zero_f32<<<dim3((2 * O + 255) / 256), 256, 0, st>>>(stats, 2 * O);
  if (O % 64 == 0) {
    dim3 grid((rows + 15) / 16, O / 64);
    gemm_bn_stats<4><<<grid, 32, 0, st>>>(X, ldx, W, K, b, Y, ldy, rows, KP, O,
                                          stats);
  } else {
    dim3 grid((rows + 15) / 16, (O + 15) / 16);
    gemm_bn_stats<1><<<grid, 32, 0, st>>>(X, ldx, W, K, b, Y, ldy, rows, KP, O,
                                          stats);
  }
  bn_finalize<<<dim3((O + 63) / 64), 64, 0, st>>>(stats, g, be, ss, O,
                                                  1.0f / (float)rows);
  size_t tot = (size_t)rows * ldy;
  bn_apply_relu<<<dim3((unsigned)((tot + 255) / 256)), 256, 0, st>>>(Y, ldy,
                                                                     rows, O,
                                                                     ss);
}

static void run_sa(hipStream_t st, const SaP& P, Bufs& W, const float* xyzCL,
                   int n, const float* feat, int Cf, int S, int Ks, float bw,
                   bool gall, float* nxOut, float* featOut, int Cout) {
  int Cin = Cf + 3;
  int KPm = ru4(Cin);
  int rows = BATCH * S * Ks;

  density_kernel<<<dim3(n / 64, BATCH), 128, 0, st>>>(xyzCL, W.invd, n, bw);

  const int* nn = nullptr;
  const float* nx = nullptr;
  if (!gall) {
    fps_kernel<<<dim3(BATCH), 256, 0, st>>>(xyzCL, n, S, W.fpsidx, nxOut);
    knn_kernel<<<dim3(S, BATCH), 32, 0, st>>>(xyzCL, nxOut, n, S, Ks, W.nnidx);
    nn = W.nnidx;
    nx = nxOut;
  }
  group_kernel<<<dim3(S, BATCH), Ks, 0, st>>>(xyzCL, feat, Cf, W.invd, nx, nn,
                                              n, S, Ks, W.Xmlp, KPm, W.Xw,
                                              W.Xds);
  // point MLP (single layer Cin -> Cout)
  mlp_layer(st, W.Xmlp, KPm, Cin, P.mlp_w, P.mlp_b, P.mlp_g, P.mlp_be, W.H,
            Cout, rows, W.stats, W.ss);
  // weightnet: 3 -> 8 -> 8 -> 16
  mlp_layer(st, W.Xw, 4, 3, P.wn_w[0], P.wn_b[0], P.wn_g[0], P.wn_be[0], W.Wb1,
            8, rows, W.stats, W.ss);
  mlp_layer(st, W.Wb1, 8, 8, P.wn_w[1], P.wn_b[1], P.wn_g[1], P.wn_be[1],
            W.Wb2, 8, rows, W.stats, W.ss);
  mlp_layer(st, W.Wb2, 8, 8, P.wn_w[2], P.wn_b[2], P.wn_g[2], P.wn_be[2],
            W.WbF, 16, rows, W.stats, W.ss);
  // densitynet: 1 -> 16 -> 8 -> 1
  mlp_layer(st, W.Xds, 4, 1, P.dn_w[0], P.dn_b[0], P.dn_g[0], P.dn_be[0],
            W.Db1, 16, rows, W.stats, W.ss);
  mlp_layer(st, W.Db1, 16, 16, P.dn_w[1], P.dn_b[1], P.dn_g[1], P.dn_be[1],
            W.Db2, 8, rows, W.stats, W.ss);
  mlp_layer(st, W.Db2, 8, 8, P.dn_w[2], P.dn_b[2], P.dn_g[2], P.dn_be[2],
            W.DbF, 1, rows, W.stats, W.ss);
  // pooled[bs, c*16+w] = sum_k (h*ds)[k,c] * wn[k,w]
  pool_kernel<<<dim3(BATCH * S, Cout / 64), 32, 0, st>>>(
      W.H, ru4(Cout), W.WbF, 16, W.DbF, Ks, Cout, W.pooled);
  // lin: (B*S, 16*Cout) -> Cout, BN over (0,1) + relu
  mlp_layer(st, W.pooled, Cout * 16, Cout * 16, P.lin_w, P.lin_b, P.lin_g,
            P.lin_be, featOut, Cout, BATCH * S, W.stats, W.ss);
}

extern "C" void kernel_launch(void* const* d_in, const int* in_sizes, int n_in,
                              void* d_out, int out_size, void* d_ws,
                              size_t ws_size, hipStream_t stream) {
  (void)in_sizes; (void)n_in; (void)out_size; (void)ws_size;
  const float* xyz = (const float*)d_in[0];
  const float* xyzg = (const float*)d_in[1];

  int pi = 2;
  auto nextp = [&]() { return (const float*)d_in[pi++]; };
  SaP SA[3];
  for (int i = 0; i < 3; ++i) {
    SA[i].mlp_w = nextp(); SA[i].mlp_b = nextp();
    SA[i].mlp_g = nextp(); SA[i].mlp_be = nextp();
    for (int l = 0; l < 3; ++l) {
      SA[i].wn_w[l] = nextp(); SA[i].wn_b[l] = nextp();
      SA[i].wn_g[l] = nextp(); SA[i].wn_be[l] = nextp();
    }
    for (int l = 0; l < 3; ++l) {
      SA[i].dn_w[l] = nextp(); SA[i].dn_b[l] = nextp();
      SA[i].dn_g[l] = nextp(); SA[i].dn_be[l] = nextp();
    }
    SA[i].lin_w = nextp(); SA[i].lin_b = nextp();
    SA[i].lin_g = nextp(); SA[i].lin_be = nextp();
  }
  const float* fc1_w = nextp(); const float* fc1_b = nextp();
  const float* gn1_g = nextp(); const float* gn1_b = nextp();
  const float* fc3_w = nextp(); const float* fc3_b = nextp();
  const float* gn3_g = nextp(); const float* gn3_b = nextp();
  const float* fc4_w = nextp(); const float* fc4_b = nextp();
  const float* gn4_g = nextp(); const float* gn4_b = nextp();
  const float* fc5_w = nextp(); const float* fc5_b = nextp();

  // ---- workspace carve (bump allocator, 256B aligned) ----
  char* base = (char*)d_ws;
  size_t off = 0;
  auto alloc = [&](size_t bytes) -> void* {
    void* p = base + off;
    off = (off + bytes + 255) & ~(size_t)255;
    return p;
  };
  const size_t RMAX = 131072;  // max rows = B*512*32
  Bufs Wb;
  Wb.pts0 = (float*)alloc(sizeof(float) * BATCH * NPTS * 3);
  Wb.nx1 = (float*)alloc(sizeof(float) * BATCH * 512 * 3);
  Wb.nx2 = (float*)alloc(sizeof(float) * BATCH * 128 * 3);
  Wb.invd = (float*)alloc(sizeof(float) * BATCH * NPTS);
  Wb.Xmlp = (float*)alloc(sizeof(float) * 65536 * 68);  // max: L2 rows x KPm
  Wb.Xw = (float*)alloc(sizeof(float) * RMAX * 4);
  Wb.Xds = (float*)alloc(sizeof(float) * RMAX * 4);
  Wb.H = (float*)alloc(sizeof(float) * RMAX * 64);  // == 65536*128
  Wb.Wb1 = (float*)alloc(sizeof(float) * RMAX * 8);
  Wb.Wb2 = (float*)alloc(sizeof(float) * RMAX * 8);
  Wb.WbF = (float*)alloc(sizeof(float) * RMAX * 16);
  Wb.Db1 = (float*)alloc(sizeof(float) * RMAX * 16);
  Wb.Db2 = (float*)alloc(sizeof(float) * RMAX * 8);
  Wb.DbF = (float*)alloc(sizeof(float) * RMAX * 4);
  Wb.pooled = (float*)alloc(sizeof(float) * 4096 * 1024);
  Wb.f1 = (float*)alloc(sizeof(float) * BATCH * 512 * 64);
  Wb.f2 = (float*)alloc(sizeof(float) * BATCH * 128 * 128);
  Wb.f3a = (float*)alloc(sizeof(float) * BATCH * 256);
  Wb.f3b = (float*)alloc(sizeof(float) * BATCH * 256);
  Wb.stats = (float*)alloc(sizeof(float) * 1024);
  Wb.ss = (float*)alloc(sizeof(float) * 1024);
  Wb.fpsidx = (int*)alloc(sizeof(int) * BATCH * 512);
  Wb.nnidx = (int*)alloc(sizeof(int) * RMAX);

  for (int e = 0; e < 2; ++e) {
    const float* inx = e ? xyzg : xyz;
    float* f3 = e ? Wb.f3b : Wb.f3a;
    to_chlast<<<dim3((NPTS + 255) / 256, BATCH), 256, 0, stream>>>(inx,
                                                                   Wb.pts0,
                                                                   NPTS);
    // SA1: n=2048, feat=pts(3), S=512, K=32, bw=0.1, Cout=64
    run_sa(stream, SA[0], Wb, Wb.pts0, 2048, Wb.pts0, 3, 512, 32, 0.1f, false,
           Wb.nx1, Wb.f1, 64);
    // SA2: n=512, feat=f1(64), S=128, K=64, bw=0.2, Cout=128
    run_sa(stream, SA[1], Wb, Wb.nx1, 512, Wb.f1, 64, 128, 64, 0.2f, false,
           Wb.nx2, Wb.f2, 128);
    // SA3: group_all, n=128, feat=f2(128), S=1, K=128, bw=0.4, Cout=256
    run_sa(stream, SA[2], Wb, Wb.nx2, 128, Wb.f2, 128, 1, 128, 0.4f, true,
           nullptr, f3, 256);
  }
  head_kernel<<<dim3(BATCH), 256, 0, stream>>>(
      Wb.f3a, Wb.f3b, fc1_w, fc1_b, gn1_g, gn1_b, fc3_w, fc3_b, gn3_g, gn3_b,
      fc4_w, fc4_b, gn4_g, gn4_b, fc5_w, fc5_b, (float*)d_out);
}